// PerPointAttention_15212774163083
// MI455X (gfx1250) — compile-verified
//
#include <hip/hip_runtime.h>
#include <hip/hip_bf16.h>
#include <math.h>

typedef __attribute__((ext_vector_type(2))) float v2f;
typedef __attribute__((ext_vector_type(8))) float v8f;
typedef __attribute__((ext_vector_type(4))) unsigned int v4u;
typedef __attribute__((ext_vector_type(8))) int v8i;
typedef __attribute__((ext_vector_type(4))) int v4i;
typedef __attribute__((address_space(3))) float lds_float;

// Problem constants
#define NB 256   // batches
#define NK 196   // patches (keys)
#define NE 1024  // embed dim
#define NA 128   // attn dim
#define NP 52    // query points
#define SCALE_F 11.313708498984761f  // sqrt(128)

#define EC 32          // E-chunk staged in LDS for kv projection
#define LDSW 36        // padded LDS row stride (floats) == TDM pad 4 DW / 32 DW
#define ROWS_PAD 208   // 13 * 16 (196 padded)
#define CHUNK_FLOATS (ROWS_PAD * LDSW)

__device__ inline v8f vzero8() {
  v8f z;
  #pragma unroll
  for (int j = 0; j < 8; ++j) z[j] = 0.0f;
  return z;
}

__device__ inline v8f wmma4(v2f a, v2f b, v8f c) {
  // D = A(16x4 f32) * B(4x16 f32) + C(16x16 f32)
  return __builtin_amdgcn_wmma_f32_16x16x4_f32(
      /*neg_a=*/false, a, /*neg_b=*/false, b,
      /*c_mod=*/(short)0, c, /*reuse_a=*/false, /*reuse_b=*/false);
}

// TDM: async-DMA a 196x32 f32 tile (row stride NE) into LDS with a 4-DWORD pad
// inserted every 32 DWORDs (-> 36-float padded LDS rows). 2D descriptor,
// groups 2/3 unused (zero). Tracked by TENSORcnt.
__device__ inline void tdm_load_chunk(const float* gsrc, unsigned lds_byte_off) {
  const unsigned long long ga = (unsigned long long)(uintptr_t)gsrc;
  v4u g0;
  g0[0] = 1u;                                   // count=1 (valid), no gather
  g0[1] = lds_byte_off;                         // lds_addr (bytes)
  g0[2] = (unsigned)(ga & 0xFFFFFFFFu);         // global_addr[31:0]
  g0[3] = (unsigned)((ga >> 32) & 0x1FFFFFFu)   // global_addr[56:32]
          | (2u << 30);                         // type = 2 ("image")
  v8i g1;
  g1[0] = (int)((2u << 16)      // data_size = 4 bytes
                | (1u << 20)    // pad_enable
                | (4u << 22)    // pad_interval: 32 DWORDs
                | (3u << 25));  // pad_amount:   4 DWORDs
  g1[1] = (int)(((unsigned)NE & 0xFFFFu) << 16);  // tensor_dim0 = 1024 (lo16)
  g1[2] = (int)(((unsigned)NK & 0xFFFFu) << 16);  // tensor_dim1 = 196  (lo16)
  g1[3] = (int)(((unsigned)EC & 0xFFFFu) << 16);  // tile_dim0   = 32
  g1[4] = (int)NK;                                // tile_dim1   = 196
  g1[5] = (int)NE;                                // tensor_dim0_stride = 1024
  g1[6] = 0;
  g1[7] = 0;
  v4i z4;
  z4[0] = 0; z4[1] = 0; z4[2] = 0; z4[3] = 0;
  v8i z8;
  #pragma unroll
  for (int j = 0; j < 8; ++j) z8[j] = 0;
  __builtin_amdgcn_tensor_load_to_lds(g0, g1, z4, z4, z8, /*cpol=*/0);
}

// ---------------------------------------------------------------------------
// Kernel 1: keys = patches[b] @ Wk^T + patch_pos ; vals = patches[b] @ Wv^T
// grid: (NB, 2)  block: 256 (8 waves). Wave w owns output column tile tn=w.
// Patch tiles are staged by the Tensor Data Mover, double-buffered in LDS.
// ---------------------------------------------------------------------------
__global__ __launch_bounds__(256) void kv_proj_kernel(
    const float* __restrict__ patches, const float* __restrict__ Wk,
    const float* __restrict__ Wv, const float* __restrict__ patch_pos,
    float* __restrict__ keys, float* __restrict__ vals) {
  __shared__ float Pch[2 * CHUNK_FLOATS];  // 2 * 208*36*4 = 59,904 B

  const int b   = blockIdx.x;
  const int mat = blockIdx.y;  // 0 = keys (Wk, +pos), 1 = vals (Wv)
  const float* __restrict__ W = mat ? Wv : Wk;
  float* __restrict__ outp    = mat ? vals : keys;

  const int tid  = threadIdx.x;
  const int wave = tid >> 5;
  const int lane = tid & 31;
  const int half = lane >> 4;  // selects K pair
  const int n    = lane & 15;  // M for A-gather, N for B/C/D
  const int tn   = wave;       // column tile (8 tiles of 16 cover A=128)

  const unsigned lds_base = (unsigned)(uintptr_t)(lds_float*)&Pch[0];
  const float* gbase = patches + (size_t)b * NK * NE;

  // Zero the pad rows (196..207) of both buffers; TDM writes rows 0..195 only.
  for (int idx = tid; idx < 2 * (ROWS_PAD - NK) * LDSW; idx += 256) {
    const int bsel = idx / ((ROWS_PAD - NK) * LDSW);
    const int r    = idx % ((ROWS_PAD - NK) * LDSW);
    Pch[bsel * CHUNK_FLOATS + NK * LDSW + r] = 0.0f;
  }

  v8f acc[13];
  #pragma unroll
  for (int t = 0; t < 13; ++t) acc[t] = vzero8();

  constexpr int NCHUNK = NE / EC;  // 32
  if (wave == 0) tdm_load_chunk(gbase, lds_base);

  for (int c = 0; c < NCHUNK; ++c) {
    const int cur = c & 1;
    if (wave == 0) {
      if (c + 1 < NCHUNK) {
        tdm_load_chunk(gbase + (size_t)(c + 1) * EC,
                       lds_base + (unsigned)((cur ^ 1) * CHUNK_FLOATS * 4));
        __builtin_amdgcn_s_wait_tensorcnt((short)1);  // current chunk done
      } else {
        __builtin_amdgcn_s_wait_tensorcnt((short)0);
      }
    }
    __syncthreads();  // chunk `cur` visible to all waves

    const float* Pc = &Pch[cur * CHUNK_FLOATS];
    const int ebase = c * EC;
    for (int er = 0; er < EC; er += 4) {
      // B operand: B[k'][a'] = W[a'][ebase+er+k'], column a' = tn*16+n
      v2f bb;
      const float* wp = W + (size_t)(tn * 16 + n) * NE + (ebase + er + 2 * half);
      bb.x = wp[0];
      bb.y = wp[1];
      #pragma unroll
      for (int tm = 0; tm < 13; ++tm) {
        // A operand: row m = tm*16+n of staged patches, K pair by half
        v2f aa;
        const float* ap = &Pc[(tm * 16 + n) * LDSW + er + 2 * half];
        aa.x = ap[0];
        aa.y = ap[1];
        acc[tm] = wmma4(aa, bb, acc[tm]);
      }
    }
    __syncthreads();  // buffer free for the next TDM issue
  }

  // Epilogue: D VGPR i -> row M = i + 8*half, col N = n
  #pragma unroll
  for (int tm = 0; tm < 13; ++tm) {
    #pragma unroll
    for (int i = 0; i < 8; ++i) {
      const int k = tm * 16 + i + 8 * half;
      const int a = tn * 16 + n;
      if (k < NK) {
        float v = acc[tm][i];
        if (!mat) v += patch_pos[k * NA + a];
        outp[((size_t)b * NK + k) * NA + a] = v;
      }
    }
  }
}

// ---------------------------------------------------------------------------
// Kernel 2: logits = Q @ keys^T ; fused scale + prior + temperature; softmax
// grid: NB  block: 256 (8 waves). Writes attn weights to d_out attn section.
// ---------------------------------------------------------------------------
__global__ __launch_bounds__(256) void attn_softmax_kernel(
    const float* __restrict__ queries, const float* __restrict__ keys,
    const float* __restrict__ prior, const float* __restrict__ temp_p,
    float* __restrict__ attn_out) {
  __shared__ float Lg[NP * NK];  // 52*196*4 = 40,768 B

  const int b    = blockIdx.x;
  const int tid  = threadIdx.x;
  const int wave = tid >> 5;
  const int lane = tid & 31;
  const int half = lane >> 4;
  const int n    = lane & 15;

  const float t      = temp_p[0];
  const float temp   = log1pf(expf(t)) + 0.5f;  // softplus + 0.5
  const float inv_st = 1.0f / (SCALE_F * temp);
  const float inv_t  = 1.0f / temp;

  // 4 (M=52 padded to 64) x 13 (N=196 padded to 208) = 52 tiles
  for (int tidx = wave; tidx < 52; tidx += 8) {
    const int tm = tidx & 3;
    const int tn = tidx >> 2;
    v8f acc = vzero8();
    for (int a0 = 0; a0 < NA; a0 += 4) {
      v2f aa;
      aa.x = 0.f; aa.y = 0.f;
      const int q = tm * 16 + n;
      if (q < NP) {
        const float* ap = queries + (size_t)q * NA + a0 + 2 * half;
        aa.x = ap[0];
        aa.y = ap[1];
      }
      v2f bb;
      bb.x = 0.f; bb.y = 0.f;
      const int kc = tn * 16 + n;
      if (kc < NK) {
        const float* bp = keys + ((size_t)b * NK + kc) * NA + a0 + 2 * half;
        bb.x = bp[0];
        bb.y = bp[1];
      }
      acc = wmma4(aa, bb, acc);
    }
    #pragma unroll
    for (int i = 0; i < 8; ++i) {
      const int q = tm * 16 + i + 8 * half;
      const int k = tn * 16 + n;
      if (q < NP && k < NK) {
        Lg[q * NK + k] = acc[i] * inv_st + prior[q * NK + k] * inv_t;
      }
    }
  }
  __syncthreads();

  // Row softmax over k (one wave per row, wave32 shfl reductions)
  for (int q = wave; q < NP; q += 8) {
    float mx = -3.0e38f;
    for (int k = lane; k < NK; k += 32) mx = fmaxf(mx, Lg[q * NK + k]);
    #pragma unroll
    for (int off = 16; off > 0; off >>= 1) mx = fmaxf(mx, __shfl_xor(mx, off, 32));
    float s = 0.f;
    for (int k = lane; k < NK; k += 32) {
      const float e = expf(Lg[q * NK + k] - mx);
      Lg[q * NK + k] = e;
      s += e;
    }
    #pragma unroll
    for (int off = 16; off > 0; off >>= 1) s += __shfl_xor(s, off, 32);
    const float r = 1.0f / s;
    for (int k = lane; k < NK; k += 32) {
      attn_out[((size_t)b * NP + q) * NK + k] = Lg[q * NK + k] * r;
    }
  }
}

// ---------------------------------------------------------------------------
// Kernel 3: attended = attn @ vals ; out = attended @ Wo^T + bo
// grid: NB  block: 256 (8 waves).
// ---------------------------------------------------------------------------
__global__ __launch_bounds__(256) void attend_out_kernel(
    const float* __restrict__ attn, const float* __restrict__ vals,
    const float* __restrict__ Wo, const float* __restrict__ bo,
    float* __restrict__ out) {
  __shared__ float Att[64 * NA];  // 64*128*4 = 32,768 B (rows >= 52 are zero)

  const int b    = blockIdx.x;
  const int tid  = threadIdx.x;
  const int wave = tid >> 5;
  const int lane = tid & 31;
  const int half = lane >> 4;
  const int n    = lane & 15;

  // attended GEMM: M=52(pad 64)->4 tiles, N=128->8 tiles, K=196 (49 steps of 4)
  for (int tidx = wave; tidx < 32; tidx += 8) {
    const int tm = tidx & 3;
    const int tn = tidx >> 2;
    v8f acc = vzero8();
    for (int k0 = 0; k0 < NK; k0 += 4) {
      v2f aa;
      aa.x = 0.f; aa.y = 0.f;
      const int q = tm * 16 + n;
      if (q < NP) {
        const float* ap = attn + ((size_t)b * NP + q) * NK + k0 + 2 * half;
        aa.x = ap[0];
        aa.y = ap[1];
      }
      v2f bb;
      const int a = tn * 16 + n;
      bb.x = vals[((size_t)b * NK + k0 + 2 * half) * NA + a];
      bb.y = vals[((size_t)b * NK + k0 + 2 * half + 1) * NA + a];
      acc = wmma4(aa, bb, acc);
    }
    #pragma unroll
    for (int i = 0; i < 8; ++i) {
      const int q = tm * 16 + i + 8 * half;        // rows >=52 hold zeros
      Att[q * NA + tn * 16 + n] = acc[i];
    }
  }
  __syncthreads();

  // out GEMM: M=52(pad 64)->4 tiles, N=1024->64 tiles, K=128 (32 steps of 4)
  for (int tidx = wave * 32; tidx < wave * 32 + 32; ++tidx) {
    const int tm = tidx >> 6;
    const int te = tidx & 63;
    v8f acc = vzero8();
    for (int a0 = 0; a0 < NA; a0 += 4) {
      v2f aa;
      const float* ap = &Att[(tm * 16 + n) * NA + a0 + 2 * half];
      aa.x = ap[0];
      aa.y = ap[1];
      v2f bb;
      const float* bp = Wo + (size_t)(te * 16 + n) * NA + a0 + 2 * half;
      bb.x = bp[0];
      bb.y = bp[1];
      acc = wmma4(aa, bb, acc);
    }
    #pragma unroll
    for (int i = 0; i < 8; ++i) {
      const int q = tm * 16 + i + 8 * half;
      if (q < NP) {
        const int e = te * 16 + n;
        out[((size_t)b * NP + q) * NE + e] = acc[i] + bo[e];
      }
    }
  }
}

// ---------------------------------------------------------------------------
extern "C" void kernel_launch(void* const* d_in, const int* in_sizes, int n_in,
                              void* d_out, int out_size, void* d_ws, size_t ws_size,
                              hipStream_t stream) {
  const float* patches   = (const float*)d_in[0];  // [256,196,1024]
  const float* queries   = (const float*)d_in[1];  // [52,128]
  const float* Wk        = (const float*)d_in[2];  // [128,1024]
  const float* Wv        = (const float*)d_in[3];  // [128,1024]
  const float* Wo        = (const float*)d_in[4];  // [1024,128]
  const float* bo        = (const float*)d_in[5];  // [1024]
  const float* patch_pos = (const float*)d_in[6];  // [196,128]
  const float* temp_p    = (const float*)d_in[7];  // scalar
  const float* prior     = (const float*)d_in[8];  // [52,196]

  float* out      = (float*)d_out;                          // [256,52,1024]
  float* attn_out = (float*)d_out + (size_t)NB * NP * NE;   // [256,52,196]

  float* keys = (float*)d_ws;                               // [256,196,128]
  float* vals = keys + (size_t)NB * NK * NA;                // [256,196,128]

  kv_proj_kernel<<<dim3(NB, 2), 256, 0, stream>>>(patches, Wk, Wv, patch_pos,
                                                  keys, vals);
  attn_softmax_kernel<<<NB, 256, 0, stream>>>(queries, keys, prior, temp_p,
                                              attn_out);
  attend_out_kernel<<<NB, 256, 0, stream>>>(attn_out, vals, Wo, bo, out);
}